// kernel_generated_0_21887153341387
// MI455X (gfx1250) — compile-verified
//
#include <hip/hip_runtime.h>

typedef __attribute__((ext_vector_type(2))) float v2f;
typedef __attribute__((ext_vector_type(8))) float v8f;

#define HH     56
#define WW     56
#define NCOL   3136   // 56*56  (n,o) flattened
#define KDIM   288    // 3*32*3 (j,l,k) flattened == w's row layout
#define MDIM   64
#define KSTEPS 72     // KDIM / 4
#define NTILES 196    // NCOL / 16

// ---------------------------------------------------------------------------
// Kernel 1: im2col directly into WMMA-B-fragment-packed layout.
//   Bpack[tile][s][lane][c],  tile = n-tile (16 cols), s = k-step (K=4),
//   lane = wave lane, c = 0/1 (the two K rows this lane supplies).
// Per ISA 7.12.2 (32-bit B 4x16): lanes 0-15 carry K rows {4s, 4s+1},
// lanes 16-31 carry K rows {4s+2, 4s+3}; column = tile*16 + (lane&15).
// Value encodes: first-unfold zero pad in rows, merged-(56,2)-axis roll
// (wraps: channel l pairs with channel 32+l of row (n-1)%56), axis-4 roll
// ((w-1)%56), and second-unfold ZERO pad at o+k-1.
// ---------------------------------------------------------------------------
__global__ __launch_bounds__(256) void build_b(const float* __restrict__ x,
                                               float* __restrict__ Bpack) {
    int idx  = blockIdx.x * 256 + threadIdx.x;      // 0 .. 288*3136-1
    int c    = idx & 1;
    int lane = (idx >> 1) & 31;
    int rest = idx >> 6;                            // tile*72 + s
    int tile = rest / KSTEPS;
    int s    = rest - tile * KSTEPS;

    int half = lane >> 4;
    int krow = (s << 2) + (half << 1) + c;          // kk in [0,288)
    int ncol = (tile << 4) + (lane & 15);           // in [0,3136)

    int j    = krow / 96;
    int rem  = krow - j * 96;
    int l    = rem / 3;
    int k    = rem - l * 3;
    int n    = ncol / WW;
    int o    = ncol - n * WW;

    int   oc = o + k - 1;                           // second unfold: zero pad
    float v  = 0.0f;
    if (oc >= 0 && oc < WW) {
        int wm1 = oc - 1; if (wm1 < 0) wm1 += WW;   // roll +1 along w (wraps)
        int r1 = n + j - 1;                         // first unfold: zero pad
        if (r1 >= 0 && r1 < HH) v += x[(l * HH + r1) * WW + wm1];
        int n2 = n - 1; if (n2 < 0) n2 += HH;       // merged-axis roll wraps
        int r2 = n2 + j - 1;
        if (r2 >= 0 && r2 < HH) v += x[((32 + l) * HH + r2) * WW + wm1];
    }
    Bpack[idx] = v;
}

// ---------------------------------------------------------------------------
// Kernel 2: out[i][ncol] = sum_kk w[i][kk] * B[kk][ncol]
// via V_WMMA_F32_16X16X4_F32.  128 threads = 4 wave32s per block; wave m owns
// i-tile m (M=64 -> 4 tiles) and all four waves stream the same packed B
// (WGP$-hot).  Two accumulators break the serial D->C dependency chain so the
// matrix pipe can overlap consecutive WMMAs.
// ---------------------------------------------------------------------------
__global__ __launch_bounds__(128) void wmma_gemm(const float* __restrict__ Wt,
                                                 const float* __restrict__ Bpack,
                                                 float* __restrict__ out) {
    int lane = threadIdx.x & 31;
    int i0   = (threadIdx.x >> 5) << 4;   // wave id * 16
    int n0   = blockIdx.x << 4;

    int half = lane >> 4;                 // 0: lanes 0-15, 1: lanes 16-31
    int l15  = lane & 15;

    // A (16x4 f32, ISA 7.12.2): lane row = i0+l15; lanes>=16 take K=2,3.
    const float* aptr = Wt + (i0 + l15) * KDIM + (half << 1);
    // B: packed fragments, one b64 per lane per step, fully coalesced.
    const v2f*   bptr = (const v2f*)(Bpack + ((size_t)blockIdx.x * KSTEPS) * 64) + lane;

    v8f acc0 = {};
    v8f acc1 = {};
#pragma unroll 4
    for (int s = 0; s < KSTEPS; s += 2) {
        v2f a0, a1;
        a0.x = aptr[0]; a0.y = aptr[1];
        a1.x = aptr[4]; a1.y = aptr[5];
        v2f b0 = bptr[0];
        v2f b1 = bptr[32];
        acc0 = __builtin_amdgcn_wmma_f32_16x16x4_f32(
            false, a0, false, b0, (short)0, acc0, false, false);
        acc1 = __builtin_amdgcn_wmma_f32_16x16x4_f32(
            false, a1, false, b1, (short)0, acc1, false, false);
        aptr += 8;
        bptr += 64;
    }

    // C/D (16x16 f32): VGPR v = row i0+v (lanes 0-15) / i0+8+v (lanes 16-31).
    float* optr = out + (i0 + (half << 3)) * NCOL + (n0 + l15);
#pragma unroll
    for (int v = 0; v < 8; ++v) optr[v * NCOL] = acc0[v] + acc1[v];
}

// ---------------------------------------------------------------------------
extern "C" void kernel_launch(void* const* d_in, const int* in_sizes, int n_in,
                              void* d_out, int out_size, void* d_ws, size_t ws_size,
                              hipStream_t stream) {
    (void)in_sizes; (void)n_in; (void)out_size; (void)ws_size;
    const float* x  = (const float*)d_in[0];     // (1,64,56,56) fp32
    const float* w  = (const float*)d_in[1];     // (64,3,32,3)  fp32 == A[64][288]
    float* out      = (float*)d_out;             // (1,64,56,56) fp32
    float* Bpack    = (float*)d_ws;              // needs 288*3136*4 = 3.45 MiB

    build_b<<<dim3((KDIM * NCOL) / 256), 256, 0, stream>>>(x, Bpack);
    wmma_gemm<<<dim3(NTILES), 128, 0, stream>>>(w, Bpack, out);
}